// GCNClassifier_32195074851336
// MI455X (gfx1250) — compile-verified
//
#include <hip/hip_runtime.h>
#include <math.h>

typedef float v2f __attribute__((ext_vector_type(2)));
typedef float v8f __attribute__((ext_vector_type(8)));

#define NN 50000
#define NE 800000
#define DIN 256
#define DH 128
#define SPITCH 132   // padded LDS row stride (floats) to avoid bank conflicts

// Native fp32 global atomic add (GLOBAL_ATOMIC_ADD_F32) — avoid CAS loops.
__device__ __forceinline__ void atomic_add_f32(float* p, float v) {
  unsafeAtomicAdd(p, v);
}

// ---------------------------------------------------------------------------
// init: zero aggregation buffer, deg = 1.0 (self-loop contribution)
// ---------------------------------------------------------------------------
__global__ void k_init(float* __restrict__ hagg, float* __restrict__ deg,
                       int nh, int nd) {
  int t = blockIdx.x * blockDim.x + threadIdx.x;
  if (t < nh) hagg[t] = 0.0f;
  if (t < nd) deg[t] = 1.0f;
}

// ---------------------------------------------------------------------------
// degree: deg[dst] += 1 for every edge
// ---------------------------------------------------------------------------
__global__ void k_degree(const long long* __restrict__ ei,
                         float* __restrict__ deg, int nedges) {
  int e = blockIdx.x * blockDim.x + threadIdx.x;
  if (e < nedges) {
    long long d = ei[nedges + e];
    atomic_add_f32(&deg[d], 1.0f);
  }
}

// ---------------------------------------------------------------------------
// deg -> dinv = 1/sqrt(deg) in place
// ---------------------------------------------------------------------------
__global__ void k_rsqrt(float* __restrict__ deg, int n) {
  int t = blockIdx.x * blockDim.x + threadIdx.x;
  if (t < n) {
    float d = deg[t];
    deg[t] = (d > 0.0f) ? (1.0f / sqrtf(d)) : 0.0f;
  }
}

// ---------------------------------------------------------------------------
// scatter: hagg[dst] += h0[src] * dinv[src]*dinv[dst]
// One wave per edge; each lane owns 4 consecutive columns (float4 load +
// 4 native f32 atomics). Coalesced 128B row accesses, indices loaded once.
// ---------------------------------------------------------------------------
__global__ void k_scatter(const float* __restrict__ h0,
                          const long long* __restrict__ ei,
                          const float* __restrict__ dinv,
                          float* __restrict__ hagg, int nedges) {
  int gtid = blockIdx.x * blockDim.x + threadIdx.x;
  int e = gtid >> 5;                    // wave-uniform edge id
  if (e >= nedges) return;
  int lane = threadIdx.x & 31;
  long long s = ei[e];
  long long d = ei[nedges + e];
  float norm = dinv[s] * dinv[d];
  int c0 = lane * 4;
  const float4 v = *(const float4*)&h0[s * DH + c0];
  float* dst = &hagg[d * DH + c0];
  atomic_add_f32(dst + 0, v.x * norm);
  atomic_add_f32(dst + 1, v.y * norm);
  atomic_add_f32(dst + 2, v.z * norm);
  atomic_add_f32(dst + 3, v.w * norm);
}

// ---------------------------------------------------------------------------
// self-loop + bias + relu: hagg = relu(hagg + h0*dinv^2 + b_gcn)  (float4)
// ---------------------------------------------------------------------------
__global__ void k_self_bias_relu(float* __restrict__ hagg,
                                 const float* __restrict__ h0,
                                 const float* __restrict__ dinv,
                                 const float* __restrict__ bgcn, int nnodes) {
  int t = blockIdx.x * blockDim.x + threadIdx.x;   // over nnodes*32 quads
  if (t >= nnodes * (DH / 4)) return;
  int i = t >> 5;                 // node
  int q = (t & 31) * 4;           // column quad
  float di = dinv[i];
  float dd = di * di;
  float4 a = *(const float4*)&hagg[i * DH + q];
  float4 h = *(const float4*)&h0[i * DH + q];
  float4 b = *(const float4*)&bgcn[q];
  float4 r;
  r.x = fmaxf(a.x + h.x * dd + b.x, 0.0f);
  r.y = fmaxf(a.y + h.y * dd + b.y, 0.0f);
  r.z = fmaxf(a.z + h.z * dd + b.z, 0.0f);
  r.w = fmaxf(a.w + h.w * dd + b.w, 0.0f);
  *(float4*)&hagg[i * DH + q] = r;
}

// ---------------------------------------------------------------------------
// WMMA f32 GEMM: C[M x 128] = op(A[M x K] @ W[K x 128] (+bias) (relu))
// One wave computes a 16x128 tile: 8 accumulators of v8f, K-loop in steps of 4
// using V_WMMA_F32_16X16X4_F32.
//   A frag (16x4):  lane m=lane&15, half=lane>>4 holds A[m][k0+2*half(+1)]
//   B frag (4x16):  vgpr r, half h holds B[k0 + r + 2h][n=lane&15 + 16*nt]
//   C tile (16x16): vgpr r, half h holds C[r + 8h][lane&15]
// ---------------------------------------------------------------------------
template <int K>
__global__ void k_gemm_wmma(const float* __restrict__ A,
                            const float* __restrict__ W,
                            const float* __restrict__ bias,
                            float* __restrict__ C, int M, int relu_en) {
  const int N = DH;
  int wave = threadIdx.x >> 5;
  int lane = threadIdx.x & 31;
  int tile = blockIdx.x * (blockDim.x >> 5) + wave;
  if (tile * 16 >= M) return;          // wave-uniform: EXEC stays all-ones
  int m = lane & 15;
  int half = lane >> 4;
  long long row = (long long)tile * 16 + m;

  v8f acc[8];
#pragma unroll
  for (int nt = 0; nt < 8; ++nt) {
    float bv = bias ? bias[nt * 16 + m] : 0.0f;
    acc[nt] = (v8f){bv, bv, bv, bv, bv, bv, bv, bv};
  }

  for (int k0 = 0; k0 < K; k0 += 4) {
    int ka = k0 + 2 * half;
    v2f a;
    a.x = A[row * K + ka];
    a.y = A[row * K + ka + 1];
#pragma unroll
    for (int nt = 0; nt < 8; ++nt) {
      int n = nt * 16 + m;
      v2f b;
      b.x = W[(long long)(k0 + 2 * half) * N + n];
      b.y = W[(long long)(k0 + 1 + 2 * half) * N + n];
      acc[nt] = __builtin_amdgcn_wmma_f32_16x16x4_f32(
          false, a, false, b, (short)0, acc[nt], false, false);
    }
  }

#pragma unroll
  for (int nt = 0; nt < 8; ++nt) {
#pragma unroll
    for (int r = 0; r < 8; ++r) {
      float v = acc[nt][r];
      if (relu_en) v = fmaxf(v, 0.0f);
      C[((long long)tile * 16 + r + 8 * half) * N + nt * 16 + m] = v;
    }
  }
}

// ---------------------------------------------------------------------------
// Fused edge kernel: gather (h[src]+h[dst])/2 -> fc1+relu -> fc2+relu ->
// 128x2 head -> log_softmax -> out. One wave owns 16 edges.
// ---------------------------------------------------------------------------
__global__ void __launch_bounds__(128)
k_edge_mlp(const float* __restrict__ h, const long long* __restrict__ ei,
           const float* __restrict__ f1W, const float* __restrict__ f1b,
           const float* __restrict__ f2W, const float* __restrict__ f2b,
           const float* __restrict__ oW, const float* __restrict__ ob,
           float* __restrict__ out, int nedges) {
  __shared__ float stag[4][16 * SPITCH];
  int wave = threadIdx.x >> 5;
  int lane = threadIdx.x & 31;
  int tile = blockIdx.x * 4 + wave;       // grid sized so every wave is valid
  int m = lane & 15;
  int half = lane >> 4;
  float* S = stag[wave];

  // gather 16 edges x 128 features into LDS
  for (int e = 0; e < 16; ++e) {
    long long s = ei[(long long)tile * 16 + e];
    long long d = ei[(long long)nedges + tile * 16 + e];
#pragma unroll
    for (int i = 0; i < 4; ++i) {
      int c = lane + 32 * i;
      S[e * SPITCH + c] = 0.5f * (h[s * DH + c] + h[d * DH + c]);
    }
  }
  __syncthreads();

  v8f acc[8];
  for (int layer = 0; layer < 2; ++layer) {
    const float* W = layer ? f2W : f1W;
    const float* B = layer ? f2b : f1b;
#pragma unroll
    for (int nt = 0; nt < 8; ++nt) {
      float bv = B[nt * 16 + m];
      acc[nt] = (v8f){bv, bv, bv, bv, bv, bv, bv, bv};
    }
    for (int k0 = 0; k0 < DH; k0 += 4) {
      int ka = k0 + 2 * half;
      v2f a;
      a.x = S[m * SPITCH + ka];
      a.y = S[m * SPITCH + ka + 1];
#pragma unroll
      for (int nt = 0; nt < 8; ++nt) {
        int n = nt * 16 + m;
        v2f b;
        b.x = W[(k0 + 2 * half) * DH + n];
        b.y = W[(k0 + 1 + 2 * half) * DH + n];
        acc[nt] = __builtin_amdgcn_wmma_f32_16x16x4_f32(
            false, a, false, b, (short)0, acc[nt], false, false);
      }
    }
    __syncthreads();
    // write relu(acc) back to LDS (C layout -> row major)
#pragma unroll
    for (int nt = 0; nt < 8; ++nt) {
#pragma unroll
      for (int r = 0; r < 8; ++r) {
        S[(r + 8 * half) * SPITCH + nt * 16 + m] = fmaxf(acc[nt][r], 0.0f);
      }
    }
    __syncthreads();
  }

  // 128 -> 2 head + log_softmax; lanes 0..15 each own one edge of the tile
  if (lane < 16) {
    float s0 = ob[0];
    float s1 = ob[1];
    for (int c = 0; c < DH; ++c) {
      float v = S[m * SPITCH + c];
      s0 += v * oW[c * 2 + 0];
      s1 += v * oW[c * 2 + 1];
    }
    float mx = fmaxf(s0, s1);
    float lse = mx + logf(expf(s0 - mx) + expf(s1 - mx));
    long long eidx = (long long)tile * 16 + m;
    out[eidx * 2 + 0] = s0 - lse;
    out[eidx * 2 + 1] = s1 - lse;
  }
}

// ---------------------------------------------------------------------------
// launch
// ---------------------------------------------------------------------------
extern "C" void kernel_launch(void* const* d_in, const int* in_sizes, int n_in,
                              void* d_out, int out_size, void* d_ws,
                              size_t ws_size, hipStream_t stream) {
  const float* x       = (const float*)d_in[0];
  const long long* ei  = (const long long*)d_in[1];
  const float* Wg      = (const float*)d_in[2];
  const float* bg      = (const float*)d_in[3];
  const float* f1W     = (const float*)d_in[4];
  const float* f1b     = (const float*)d_in[5];
  const float* f2W     = (const float*)d_in[6];
  const float* f2b     = (const float*)d_in[7];
  const float* oW      = (const float*)d_in[8];
  const float* ob      = (const float*)d_in[9];
  float* out           = (float*)d_out;

  // workspace layout: h0 | hbuf | deg   (~51.4 MB)
  float* h0   = (float*)d_ws;
  float* hbuf = h0 + (size_t)NN * DH;
  float* deg  = hbuf + (size_t)NN * DH;

  const int nh = NN * DH;               // 6,400,000

  // 1) zero aggregation buffer, deg = 1 (self loops)
  k_init<<<(nh + 255) / 256, 256, 0, stream>>>(hbuf, deg, nh, NN);
  // 2) degree accumulation
  k_degree<<<(NE + 255) / 256, 256, 0, stream>>>(ei, deg, NE);
  // 3) h0 = x @ W_gcn (WMMA f32)
  {
    int tiles = NN / 16;                // 3125 exact
    int blocks = (tiles + 3) / 4;
    k_gemm_wmma<DIN><<<blocks, 128, 0, stream>>>(x, Wg, nullptr, h0, NN, 0);
  }
  // 4) dinv = 1/sqrt(deg)
  k_rsqrt<<<(NN + 255) / 256, 256, 0, stream>>>(deg, NN);
  // 5) normalized scatter-add: one wave per edge
  {
    long long threads = (long long)NE * 32;
    int blocks = (int)((threads + 255) / 256);   // 100000
    k_scatter<<<blocks, 256, 0, stream>>>(h0, ei, deg, hbuf, NE);
  }
  // 6) add self-loop term + bias, relu (float4)
  k_self_bias_relu<<<(NN * (DH / 4) + 255) / 256, 256, 0, stream>>>(hbuf, h0,
                                                                    deg, bg,
                                                                    NN);
  // 7) node MLP: h0 = relu(hbuf@fc1+b1); hbuf = relu(h0@fc2+b2)
  {
    int tiles = NN / 16;
    int blocks = (tiles + 3) / 4;
    k_gemm_wmma<DH><<<blocks, 128, 0, stream>>>(hbuf, f1W, f1b, h0, NN, 1);
    k_gemm_wmma<DH><<<blocks, 128, 0, stream>>>(h0, f2W, f2b, hbuf, NN, 1);
  }
  // 8) fused edge gather + MLP + head + log_softmax
  {
    int tiles = NE / 16;                // 50000 exact
    int blocks = tiles / 4;             // 12500, every wave valid
    k_edge_mlp<<<blocks, 128, 0, stream>>>(hbuf, ei, f1W, f1b, f2W, f2b, oW,
                                           ob, out, NE);
  }
}